// _ChaosEncoder_37426345017639
// MI455X (gfx1250) — compile-verified
//
#include <hip/hip_runtime.h>
#include <math.h>

typedef float v2f __attribute__((ext_vector_type(2)));
typedef float v4f __attribute__((ext_vector_type(4)));
typedef float v8f __attribute__((ext_vector_type(8)));

#define ALPHA   2.1f
#define INV2PI  0.15915494309189535f
#define CREV    (ALPHA * INV2PI)     // v_cos_f32 computes cos(2*pi*x): pre-scale to revolutions
#define NFOLDS  64
#define DFEAT   512                  // feature dim (also K of the GEMM)
#define ROWS    16                   // rows (b,m pairs) per workgroup
#define THREADS 256                  // 8 wave32 waves
#define EPT     ((ROWS * DFEAT) / THREADS)   // 32 contiguous elements per thread (1/16 of a row)

__device__ __forceinline__ float lane_bcast(int srcLane, float v) {
    return __int_as_float(__builtin_amdgcn_ds_bpermute(srcLane << 2, __float_as_int(v)));
}
__device__ __forceinline__ float cos2pi(float x) {        // cos(2*pi*x) -> v_cos_f32
    return __builtin_amdgcn_cosf(x);
}

__global__ __launch_bounds__(THREADS) void chaos_encoder_kernel(
    const float* __restrict__ x, const float* __restrict__ P,
    float* __restrict__ out)
{
    __shared__ __align__(16) float regionA[ROWS * DFEAT];   // x tile transposed (GEMM only)
    __shared__ __align__(16) float regionB[ROWS * DFEAT];   // E tile (WMMA output)

    const int t    = threadIdx.x;
    const int lane = t & 31;
    const int wave = t >> 5;
    const size_t tileBase = (size_t)blockIdx.x * (ROWS * DFEAT);

    // ---------------- stage x tile transposed: sXT[k*16 + m] = x[m][k] ----------------
    float* sXT = regionA;
    #pragma unroll
    for (int j = 0; j < EPT; ++j) {
        int e = t + THREADS * j;          // e = m*512 + k  (coalesced global read)
        int m = e >> 9;
        int k = e & (DFEAT - 1);
        sXT[k * ROWS + m] = x[tileBase + e];
    }
    __syncthreads();

    // ---------------- GEMM: E(16x512) = xTile(16x512) @ P^T via V_WMMA_F32_16X16X4_F32 ----
    // A (16x4 fp32): lanes 0-15 -> M=lane, VGPR0/1 = K0/K1 ; lanes 16-31 -> M=lane-16, K2/K3
    // B (4x16 fp32): VGPR0 lanes 0-15 = K0 row, lanes 16-31 = K2 row; VGPR1 = K1/K3 rows
    const int half = lane >> 4;
    const int sub  = lane & 15;
    const int d0   = wave * 64 + sub;
    const float* p0 = P + (size_t)(d0     ) * DFEAT;
    const float* p1 = P + (size_t)(d0 + 16) * DFEAT;
    const float* p2 = P + (size_t)(d0 + 32) * DFEAT;
    const float* p3 = P + (size_t)(d0 + 48) * DFEAT;

    v8f acc0 = {}, acc1 = {}, acc2 = {}, acc3 = {};
    for (int k0 = 0; k0 < DFEAT; k0 += 4) {
        const int kk = k0 + 2 * half;
        v2f a;
        a.x = sXT[(kk    ) * ROWS + sub];
        a.y = sXT[(kk + 1) * ROWS + sub];
        v2f b0 = *(const v2f*)(p0 + kk);
        v2f b1 = *(const v2f*)(p1 + kk);
        v2f b2 = *(const v2f*)(p2 + kk);
        v2f b3 = *(const v2f*)(p3 + kk);
        acc0 = __builtin_amdgcn_wmma_f32_16x16x4_f32(false, a, false, b0, (short)0, acc0, false, false);
        acc1 = __builtin_amdgcn_wmma_f32_16x16x4_f32(false, a, false, b1, (short)0, acc1, false, false);
        acc2 = __builtin_amdgcn_wmma_f32_16x16x4_f32(false, a, false, b2, (short)0, acc2, false, false);
        acc3 = __builtin_amdgcn_wmma_f32_16x16x4_f32(false, a, false, b3, (short)0, acc3, false, false);
    }

    // scatter accumulators into E tile, C/D layout: M = v + 8*half, N = sub
    float* sE = regionB;
    {
        const int col = wave * 64 + sub;
        #pragma unroll
        for (int v = 0; v < 8; ++v) {
            const int m = v + 8 * half;
            sE[m * DFEAT + col +  0] = acc0[v];
            sE[m * DFEAT + col + 16] = acc1[v];
            sE[m * DFEAT + col + 32] = acc2[v];
            sE[m * DFEAT + col + 48] = acc3[v];
        }
    }
    __syncthreads();

    // ---------------- pull E into chunk-ordered registers, pre-scaled to revolutions ------
    // thread t owns contiguous elements [t*32, t*32+32) -> one 16-lane group per row
    float ce[EPT];
    {
        const v4f* sEv = (const v4f*)(sE + t * EPT);
        #pragma unroll
        for (int q = 0; q < EPT / 4; ++q) {
            v4f v = sEv[q];
            ce[4*q + 0] = v.x * CREV;
            ce[4*q + 1] = v.y * CREV;
            ce[4*q + 2] = v.z * CREV;
            ce[4*q + 3] = v.w * CREV;
        }
    }

    // ---------------- chaotic fold loop: fully register-resident, no barriers -------------
    // roll(+1) source of element 0 of chunk c is element 31 of chunk (c-1) mod 16,
    // i.e. a lane-rotate within each 16-lane group: one ds_bpermute per fold.
    const int srcLane = (lane & 16) | ((lane + 15) & 15);

    float s[EPT];
    #pragma unroll
    for (int i = 0; i < EPT; ++i)        // fold 0: previous state is zero
        s[i] = cos2pi(ce[i]);

    #pragma unroll 1
    for (int it = 1; it < NFOLDS; ++it) {
        const float prev = lane_bcast(srcLane, s[EPT - 1]);   // neighbor chunk's old last elem
        #pragma unroll
        for (int i = EPT - 1; i >= 1; --i)                    // in-place, descending: uses old s[i-1]
            s[i] = cos2pi(fmaf(CREV, s[i - 1], ce[i]));
        s[0] = cos2pi(fmaf(CREV, prev, ce[0]));
    }

    // ---------------- per-row L2 norm: 16-lane xor-butterfly, then fused scale + store ----
    float ss = 0.f;
    #pragma unroll
    for (int i = 0; i < EPT; ++i) ss += s[i] * s[i];
    #pragma unroll
    for (int m = 8; m >= 1; m >>= 1)
        ss += lane_bcast(lane ^ m, ss);                       // stays within the 16-lane group
    const float inv = 1.0f / fmaxf(sqrtf(ss), 1e-8f);

    v4f* op = (v4f*)(out + tileBase + (size_t)t * EPT);       // 128B-aligned, coalesced b128
    #pragma unroll
    for (int q = 0; q < EPT / 4; ++q) {
        v4f v;
        v.x = s[4*q + 0] * inv;
        v.y = s[4*q + 1] * inv;
        v.z = s[4*q + 2] * inv;
        v.w = s[4*q + 3] * inv;
        op[q] = v;
    }
}

extern "C" void kernel_launch(void* const* d_in, const int* in_sizes, int n_in,
                              void* d_out, int out_size, void* d_ws, size_t ws_size,
                              hipStream_t stream) {
    const float* x = (const float*)d_in[0];   // (8, 4096, 512) fp32
    const float* P = (const float*)d_in[1];   // (512, 512) fp32
    float* out = (float*)d_out;               // (8, 4096, 512) fp32

    const int totalRows = in_sizes[0] / DFEAT;   // 32768
    const int blocks = totalRows / ROWS;         // 2048
    chaos_encoder_kernel<<<blocks, THREADS, 0, stream>>>(x, P, out);
}